// Render_9457517985874
// MI455X (gfx1250) — compile-verified
//
#include <hip/hip_runtime.h>

typedef float v2f __attribute__((ext_vector_type(2)));
typedef float v8f __attribute__((ext_vector_type(8)));

#define RSIZE 512
#define NTRI  512
#define NEGF  (-1e30f)

// linspace(-1,1,512)[i]
__device__ __forceinline__ float pix_coord(int i) {
    return fmaf((float)i, 2.0f / 511.0f, -1.0f);
}
// half-grid threshold between pixel bb-1 and bb:  xi >= bb  <=>  px > thresh(bb)
//                                                 xi <  bb  <=>  px < thresh(bb)
__device__ __forceinline__ float bb_thresh(int bb) {
    return fmaf((float)(2 * bb - 1), 1.0f / 511.0f, -1.0f);
}

// Per-triangle affine coefficients for 8 components:
//   0..2: edge functions pAB,pBC,pCA   3: z = (pAB*Az+pBC*Bz+pCA*Cz)/w
//   4..7: bbox tests as sign tests (px - tlo), (thi - px), (py - tlo), (thi - py)
// Laid out per group of 2 triangles exactly as the WMMA A-operand wants it:
//   coef[g*32 + l]      (l<16) = (a_m, b_m)   for row m=l   (A[K=0],A[K=1])
//   coef[g*32 + 16 + l']        = (c_m, 0)    for row m=l'  (A[K=2],A[K=3])
__global__ __launch_bounds__(256) void setup_coef(const float* __restrict__ tris,
                                                  float2* __restrict__ coef) {
    int t = blockIdx.x * blockDim.x + threadIdx.x;
    if (t >= NTRI) return;
    const float* p = tris + t * 9;
    float Ax=p[0],Ay=p[1],Az=p[2], Bx=p[3],By=p[4],Bz=p[5], Cx=p[6],Cy=p[7],Cz=p[8];
    float w = (Bx-Ax)*(Cy-Ay) - (By-Ay)*(Cx-Ax);
    // coverage requires all edges > 0 => w > 0; plus |w| > 1e-8. So cull unless w > 1e-8.
    bool ok = w > 1e-8f;
    float a[8], b[8], c[8];
    a[0]=Ax*By-Ay*Bx; b[0]=Ay-By; c[0]=Bx-Ax;      // pAB = a + b*px + c*py
    a[1]=Bx*Cy-By*Cx; b[1]=By-Cy; c[1]=Cx-Bx;      // pBC
    a[2]=Cx*Ay-Cy*Ax; b[2]=Cy-Ay; c[2]=Ax-Cx;      // pCA
    float invw = ok ? 1.0f / w : 0.0f;
    a[3]=(a[0]*Az + a[1]*Bz + a[2]*Cz)*invw;
    b[3]=(b[0]*Az + b[1]*Bz + b[2]*Cz)*invw;
    c[3]=(c[0]*Az + c[1]*Bz + c[2]*Cz)*invw;
    float lox = fminf(fminf(Ax,Bx),Cx), hix = fmaxf(fmaxf(Ax,Bx),Cx);
    float loy = fminf(fminf(Ay,By),Cy), hiy = fmaxf(fmaxf(Ay,By),Cy);
    lox = fminf(fmaxf(lox,-1.f),1.f); hix = fminf(fmaxf(hix,-1.f),1.f);
    loy = fminf(fmaxf(loy,-1.f),1.f); hiy = fminf(fmaxf(hiy,-1.f),1.f);
    int bb0 = (int)((lox+1.f)*0.5f*512.f);
    int bb1 = (int)((loy+1.f)*0.5f*512.f);
    int bb2 = (int)((hix+1.f)*0.5f*512.f);
    int bb3 = (int)((hiy+1.f)*0.5f*512.f);
    a[4] = -bb_thresh(bb0); b[4]= 1.f; c[4]= 0.f;  // px - tlo > 0
    a[5] =  bb_thresh(bb2); b[5]=-1.f; c[5]= 0.f;  // thi - px > 0
    a[6] = -bb_thresh(bb1); b[6]= 0.f; c[6]= 1.f;  // py - tlo > 0
    a[7] =  bb_thresh(bb3); b[7]= 0.f; c[7]=-1.f;  // thi - py > 0
    if (!ok) { a[0]=-1.f; b[0]=0.f; c[0]=0.f; a[3]=0.f; b[3]=0.f; c[3]=0.f; }
    int g = t >> 1, hi = t & 1;
    float2* base = coef + g * 32;
    #pragma unroll
    for (int j = 0; j < 8; ++j) {
        int m = hi * 8 + j;
        base[m]      = make_float2(a[j], b[j]);
        base[16 + m] = make_float2(c[j], 0.0f);
    }
}

// One wave = 16 pixels (fixed x column, 16 consecutive y).
// Each V_WMMA_F32_16X16X4_F32 evaluates 8 components x 2 triangles x 16 pixels;
// lane k (k<16) ends up holding all 8 comps of triangle 2g   at pixel k,
// lane k+16 holds all 8 comps of triangle 2g+1 at pixel k  -> lane-local select.
__global__ __launch_bounds__(256) void render_wmma(const float* __restrict__ tris,
                                                   const float2* __restrict__ coef,
                                                   float* __restrict__ out) {
    __shared__ float2 sc[4096];           // 32 KB staging, 2 phases of 128 groups
    const int tid  = threadIdx.x;
    const int lane = tid & 31;
    const int wave = tid >> 5;
    const int tile = blockIdx.x * 8 + wave;        // 2048 blocks * 8 waves = 16384 tiles
    const int x    = tile >> 5;
    const int y0   = (tile & 31) << 4;
    const int n    = lane & 15;
    const float px = pix_coord(x);
    const float py = pix_coord(y0 + n);
    const bool hiK = lane >= 16;

    // B operand: 4x16 {1, px, py, 0}; VGPR0 = K rows 0(lo)/2(hi), VGPR1 = rows 1(lo)/3(hi)
    v2f bmat;
    bmat.x = hiK ? py   : 1.0f;
    bmat.y = hiK ? 0.0f : px;
    v8f czero = {0.f,0.f,0.f,0.f,0.f,0.f,0.f,0.f};

    float best = NEGF;
    int   bidx = 0;

    for (int ph = 0; ph < 2; ++ph) {
        __syncthreads();
        for (int i = tid; i < 4096; i += 256) sc[i] = coef[ph * 4096 + i];
        __syncthreads();
        #pragma unroll 4
        for (int g = 0; g < 128; ++g) {
            float2 ab = sc[g * 32 + lane];
            v2f amat; amat.x = ab.x; amat.y = ab.y;
            v8f d = __builtin_amdgcn_wmma_f32_16x16x4_f32(
                false, amat, false, bmat, (short)0, czero, false, false);
            // coverage = all 7 tests > 0.  All values are finite (no NaN/Inf can
            // arise from finite affine coeffs), so "all positive" == "no sign bit
            // set in the OR of the raw bits" (exact-zero edge hits are measure-zero
            // ties under the reference's strict '>').  6 v_or + 1 int cmp, no
            // IEEE-minnum canonicalization ops.
            int ors = __float_as_int(d[0]) | __float_as_int(d[1]) |
                      __float_as_int(d[2]) | __float_as_int(d[4]) |
                      __float_as_int(d[5]) | __float_as_int(d[6]) |
                      __float_as_int(d[7]);
            int tri = (((ph << 7) | g) << 1) | (hiK ? 1 : 0);
            // uncovered pixels never update, so comparing z directly against best
            // (init NEG) is equivalent to the reference's key >= zbuf; '>=' keeps
            // the later triangle on exact z ties.
            bool take = (ors >= 0) && (d[3] >= best);
            best = take ? d[3] : best;
            bidx = take ? tri  : bidx;
        }
    }

    // merge even-triangle (lanes 0-15) and odd-triangle (lanes 16-31) candidates
    float obest = __shfl_xor(best, 16, 32);
    int   oidx  = __shfl_xor(bidx, 16, 32);
    if (obest > best || (obest == best && oidx > bidx)) { best = obest; bidx = oidx; }

    if (lane < 16) {
        float4 res = make_float4(0.f, 0.f, 0.f, 0.f);
        if (best > -1e29f) {
            const float* p = tris + bidx * 9;
            float Ax=p[0],Ay=p[1],Az=p[2], Bx=p[3],By=p[4],Bz=p[5], Cx=p[6],Cy=p[7],Cz=p[8];
            float pAB = (Ax-px)*(By-py) - (Ay-py)*(Bx-px);
            float pBC = (Bx-px)*(Cy-py) - (By-py)*(Cx-px);
            float w   = (Bx-Ax)*(Cy-Ay) - (By-Ay)*(Cx-Ax);
            float w1 = pAB / w, w2 = pBC / w, w3 = 1.0f - w1 - w2;
            res = make_float4(w1*Ax + w2*Bx + w3*Cx,
                              w1*Ay + w2*By + w3*Cy,
                              w1*Az + w2*Bz + w3*Cz, 1.0f);
        }
        ((float4*)out)[x * RSIZE + (y0 + n)] = res;
    }
}

extern "C" void kernel_launch(void* const* d_in, const int* in_sizes, int n_in,
                              void* d_out, int out_size, void* d_ws, size_t ws_size,
                              hipStream_t stream) {
    const float* tris = (const float*)d_in[0];
    float2* coef = (float2*)d_ws;                       // 64 KB of scratch
    setup_coef<<<2, 256, 0, stream>>>(tris, coef);
    render_wmma<<<2048, 256, 0, stream>>>(tris, coef, (float*)d_out);
}